// GraphTransformer2_52630529245426
// MI455X (gfx1250) — compile-verified
//
#include <hip/hip_runtime.h>
#include <math.h>

// Problem constants (reference: N=50000, E=500000, D=64, H=4, C=16, L=3)
#define NN 50000L
#define EEDG 500000L

typedef __attribute__((ext_vector_type(2))) float v2f;
typedef __attribute__((ext_vector_type(8))) float v8f;

// Packed-weight LDS layout: row (k>>1), element n*2 + (k&1), row stride 160 floats.
// -> B-frag (K,K+1) pair is one contiguous 8B ds_load_b64 per lane;
//    upper half-wave row offset 160 ≡ 32 (mod 64 banks): halves use disjoint banks.
#define WPK 160

// ---------------------------------------------------------------------------
// WMMA f32 16x16x4 (exact fp32 path; K looped 64/4)
// ---------------------------------------------------------------------------
__device__ __forceinline__ v8f wmma4(v2f a, v2f b, v8f c) {
  return __builtin_amdgcn_wmma_f32_16x16x4_f32(
      /*neg_a=*/false, a, /*neg_b=*/false, b,
      /*c_mod=*/(short)0, c, /*reuse_a=*/false, /*reuse_b=*/false);
}

// Stage a 64x64 row-major weight matrix into LDS in packed B-frag layout.
__device__ __forceinline__ void stage_weights_pk(const float* __restrict__ W,
                                                 float* dst, int tid, int nthr) {
  for (int i = tid; i < 4096; i += nthr) {
    int k = i >> 6, n = i & 63;
    dst[(k >> 1) * WPK + n * 2 + (k & 1)] = W[i];
  }
}

// One wave computes a 16x64 output tile: out[row0:row0+16, 0:64] = A @ W (+bias)(+relu)
// A-frag (32-bit A 16x4): lane l: M = l&15, VGPR pair = K+{0,1}, K-base 2*(l>>4) -> 8B load.
// B-frag (4x16):          lane l: N = l&15, packed pair from LDS       -> 8B ds_load_b64.
// C/D: vgpr r: lanes0-15 -> M=r, lanes16-31 -> M=r+8; N = lane&15.
// Partial tiles: clamped loads, guarded stores (EXEC all-ones for WMMA; barriers collective).
__device__ __forceinline__ void wave_gemm_16x64(
    const float* __restrict__ A, long aRow0, int aLd, long aMax,
    const float* Wpk, const float* __restrict__ bias, bool relu,
    float* __restrict__ out, long oRow0, int oLd, long oMax, int lane) {
  const int m   = lane & 15;
  const int grp = lane >> 4;
  const int kb  = grp << 1;
  long ar = aRow0 + m;
  if (ar > aMax - 1) ar = aMax - 1;
  if (ar < 0) ar = 0;
  const float* arow = A + ar * (long)aLd;

  v8f z = {0.f, 0.f, 0.f, 0.f, 0.f, 0.f, 0.f, 0.f};
  v8f acc[4] = {z, z, z, z};

#pragma unroll
  for (int k0 = 0; k0 < 64; k0 += 4) {
    v2f a = *(const v2f*)(arow + k0 + kb);               // (k0+kb) even, aLd even -> 8B aligned
    const float* wrow = Wpk + ((k0 + kb) >> 1) * WPK;
#pragma unroll
    for (int j = 0; j < 4; ++j) {
      v2f b = *(const v2f*)(wrow + (j * 16 + m) * 2);    // contiguous packed pair
      acc[j] = wmma4(a, b, acc[j]);
    }
  }

#pragma unroll
  for (int j = 0; j < 4; ++j) {
    float bb = bias ? bias[j * 16 + m] : 0.0f;
#pragma unroll
    for (int r = 0; r < 8; ++r) {
      long orow = oRow0 + r + grp * 8;
      if (orow < oMax) {
        float vv = acc[j][r] + bb;
        if (relu) vv = fmaxf(vv, 0.0f);
        out[orow * (long)oLd + j * 16 + m] = vv;
      }
    }
  }
}

// ---------------------------------------------------------------------------
// Generic 64->64 GEMM kernel: 128 threads = 4 waves, each wave a 16-row tile.
// Weight matrix staged once per block into LDS (packed layout).
// ---------------------------------------------------------------------------
__global__ void gemm64_kernel(const float* __restrict__ A,
                              const float* __restrict__ W,
                              const float* __restrict__ bias,
                              float* __restrict__ out, long M, int relu) {
  __shared__ __align__(16) float Wsh[32 * WPK];
  stage_weights_pk(W, Wsh, threadIdx.x, 128);
  __syncthreads();
  int lane = threadIdx.x & 31;
  long tile = (long)blockIdx.x * 4 + (threadIdx.x >> 5);
  long row0 = tile * 16;
  wave_gemm_16x64(A, row0, 64, M, Wsh, bias, relu != 0, out, row0, 64, M, lane);
}

// Fused q/k/v/skip: reuse the same A rows for 4 weight matrices (all staged in LDS)
__global__ void qkvs_kernel(const float* __restrict__ x,
                            const float* __restrict__ Wq, const float* __restrict__ bq,
                            const float* __restrict__ Wk, const float* __restrict__ bk,
                            const float* __restrict__ Wv, const float* __restrict__ bv,
                            const float* __restrict__ Wsk, const float* __restrict__ bsk,
                            float* __restrict__ q, float* __restrict__ k,
                            float* __restrict__ v, float* __restrict__ sk, long M) {
  __shared__ __align__(16) float Wsh[4][32 * WPK];
  stage_weights_pk(Wq,  Wsh[0], threadIdx.x, 128);
  stage_weights_pk(Wk,  Wsh[1], threadIdx.x, 128);
  stage_weights_pk(Wv,  Wsh[2], threadIdx.x, 128);
  stage_weights_pk(Wsk, Wsh[3], threadIdx.x, 128);
  __syncthreads();
  int lane = threadIdx.x & 31;
  long tile = (long)blockIdx.x * 4 + (threadIdx.x >> 5);
  long row0 = tile * 16;
  const float* bm[4] = {bq, bk, bv, bsk};
  float* om[4] = {q, k, v, sk};
#pragma unroll
  for (int w = 0; w < 4; ++w)
    wave_gemm_16x64(x, row0, 64, M, Wsh[w], bm[w], false, om[w], row0, 64, M, lane);
}

// Fused edge-encoder MLP: relu(ea@W1+b1)@W2+b2; weights + intermediate tile in LDS.
// Tile row stride 68: rows stay 16B aligned for v2f A-loads; half-wave A reads use
// disjoint bank classes ({0,1} vs {2,3} mod 4).
__global__ void edge_mlp_kernel(const float* __restrict__ ea,
                                const float* __restrict__ W1, const float* __restrict__ b1,
                                const float* __restrict__ W2, const float* __restrict__ b2,
                                float* __restrict__ eout, long E) {
  __shared__ __align__(16) float Wsh1[32 * WPK];
  __shared__ __align__(16) float Wsh2[32 * WPK];
  __shared__ __align__(16) float tile_sh[4][16 * 68];
  stage_weights_pk(W1, Wsh1, threadIdx.x, 128);
  stage_weights_pk(W2, Wsh2, threadIdx.x, 128);
  __syncthreads();
  int wave = threadIdx.x >> 5, lane = threadIdx.x & 31;
  long tile = (long)blockIdx.x * 4 + wave;
  long row0 = tile * 16;
  wave_gemm_16x64(ea, row0, 64, E, Wsh1, b1, true, &tile_sh[wave][0], 0, 68, 16, lane);
  // same wave produced + consumes the tile: no barrier needed
  wave_gemm_16x64(&tile_sh[wave][0], 0, 68, 16, Wsh2, b2, false, eout, row0, 64, E, lane);
}

// ---------------------------------------------------------------------------
// Relative-positional scalar: rp = ||x[row]-x[col]||, standardized (ddof=1)
// ---------------------------------------------------------------------------
__global__ void rp_kernel(const float* __restrict__ x,
                          const long long* __restrict__ rowI,
                          const long long* __restrict__ colI,
                          float* __restrict__ rp, float* __restrict__ rstats, long E) {
  long e = (long)blockIdx.x * blockDim.x + threadIdx.x;
  float val = 0.f, val2 = 0.f;
  if (e < E) {
    long r = (long)rowI[e] * 64, c = (long)colI[e] * 64;
    float s = 0.f;
#pragma unroll
    for (int i = 0; i < 64; ++i) {
      float dx = x[r + i] - x[c + i];
      s += dx * dx;
    }
    s = sqrtf(s);
    rp[e] = s;
    val = s;
    val2 = s * s;
  }
  for (int off = 16; off > 0; off >>= 1) {
    val  += __shfl_down(val, off, 32);
    val2 += __shfl_down(val2, off, 32);
  }
  if ((threadIdx.x & 31) == 0) {
    atomicAdd(&rstats[0], val);
    atomicAdd(&rstats[1], val2);
  }
}

__global__ void rp_apply_kernel(float* __restrict__ eout, const float* __restrict__ rp,
                                const float* __restrict__ rstats, long E) {
  long t = (long)blockIdx.x * blockDim.x + threadIdx.x;
  if (t >= E * 64) return;
  long e = t >> 6;
  float mean = rstats[0] / (float)E;
  float var = (rstats[1] - (float)E * mean * mean) / (float)(E - 1);
  float sd = sqrtf(fmaxf(var, 0.f));
  eout[t] += (rp[e] - mean) / (sd + 1e-6f);
}

// ---------------------------------------------------------------------------
// Attention: per-(edge,head) alpha + segment max / exp-sum / scatter
// ---------------------------------------------------------------------------
__device__ __forceinline__ void atomicMaxF(float* addr, float v) {
  unsigned* ua = (unsigned*)addr;
  unsigned old = *ua;
  while (true) {
    float f = __uint_as_float(old);
    if (f >= v) break;
    unsigned prev = atomicCAS(ua, old, __float_as_uint(v));
    if (prev == old) break;
    old = prev;
  }
}

__global__ void init_softmax_kernel(float* __restrict__ mx, float* __restrict__ ssum, long n4) {
  long i = (long)blockIdx.x * blockDim.x + threadIdx.x;
  if (i >= n4) return;
  mx[i] = -3.402823466e38f;
  ssum[i] = 0.f;
}

__global__ void alpha_kernel(const float* __restrict__ q, const float* __restrict__ k,
                             const float* __restrict__ ee,
                             const long long* __restrict__ rowI,
                             const long long* __restrict__ colI,
                             float* __restrict__ alpha, float* __restrict__ mx, long E) {
  long t = (long)blockIdx.x * blockDim.x + threadIdx.x;
  if (t >= E * 4) return;
  long e = t >> 2;
  int h = (int)(t & 3);
  long r = (long)rowI[e], c = (long)colI[e];
  const float* qp = q + c * 64 + h * 16;
  const float* kp = k + r * 64 + h * 16;
  const float* ep = ee + e * 64 + h * 16;
  float s = 0.f;
#pragma unroll
  for (int i = 0; i < 16; ++i) s += qp[i] * (kp[i] + ep[i]);
  s *= 0.25f;  // 1/sqrt(C=16)
  alpha[t] = s;
  atomicMaxF(&mx[c * 4 + h], s);
}

__global__ void expsum_kernel(const float* __restrict__ alpha,
                              const long long* __restrict__ colI,
                              const float* __restrict__ mx,
                              float* __restrict__ expa, float* __restrict__ ssum, long E) {
  long t = (long)blockIdx.x * blockDim.x + threadIdx.x;
  if (t >= E * 4) return;
  long e = t >> 2;
  int h = (int)(t & 3);
  long c = (long)colI[e];
  float a = expf(alpha[t] - mx[c * 4 + h]);
  expa[t] = a;
  atomicAdd(&ssum[c * 4 + h], a);
}

__global__ void scatter_kernel(const float* __restrict__ v, const float* __restrict__ ee,
                               const float* __restrict__ expa, const float* __restrict__ ssum,
                               const long long* __restrict__ rowI,
                               const long long* __restrict__ colI,
                               float* __restrict__ outacc, long E) {
  long t = (long)blockIdx.x * blockDim.x + threadIdx.x;
  if (t >= E * 64) return;
  long e = t >> 6;
  int d = (int)(t & 63);
  int h = d >> 4;
  long r = (long)rowI[e], c = (long)colI[e];
  float coef = expa[e * 4 + h] / (ssum[c * 4 + h] + 1e-16f);
  float val = (v[r * 64 + d] + ee[e * 64 + d]) * coef;
  atomicAdd(&outacc[c * 64 + d], val);
}

// ---------------------------------------------------------------------------
// BatchNorm stats (per-column over N) + fused BN+ReLU+residual+LayerNorm
// ---------------------------------------------------------------------------
__global__ void bnstats_kernel(const float* __restrict__ h,
                               float* __restrict__ colsum, float* __restrict__ colsq,
                               long total) {
  long stride = (long)gridDim.x * blockDim.x;  // multiple of 64 by launch config
  long tid = (long)blockIdx.x * blockDim.x + threadIdx.x;
  int d = (int)(tid & 63);
  float s = 0.f, s2 = 0.f;
  for (long i = tid; i < total; i += stride) {
    float u = h[i];
    s += u;
    s2 += u * u;
  }
  atomicAdd(&colsum[d], s);
  atomicAdd(&colsq[d], s2);
}

__global__ void bnln_kernel(const float* __restrict__ h, const float* __restrict__ xres,
                            const float* __restrict__ colsum, const float* __restrict__ colsq,
                            const float* __restrict__ bn_g, const float* __restrict__ bn_b,
                            const float* __restrict__ ln_g, const float* __restrict__ ln_b,
                            float* __restrict__ xout, long Nn) {
  long nid = blockIdx.x;
  int d = threadIdx.x;
  float invN = 1.0f / (float)Nn;
  float mu = colsum[d] * invN;
  float var = colsq[d] * invN - mu * mu;
  float hv = (h[nid * 64 + d] - mu) * rsqrtf(var + 1e-5f) * bn_g[d] + bn_b[d];
  hv = fmaxf(hv, 0.f);
  float t = hv + xres[nid * 64 + d];
  __shared__ float red[64];
  red[d] = t;
  __syncthreads();
  float s = 0.f, s2 = 0.f;
#pragma unroll
  for (int i = 0; i < 64; ++i) {
    float u = red[i];
    s += u;
    s2 += u * u;
  }
  float m2 = s * (1.0f / 64.0f);
  float vr = s2 * (1.0f / 64.0f) - m2 * m2;
  xout[nid * 64 + d] = (t - m2) * rsqrtf(vr + 1e-5f) * ln_g[d] + ln_b[d];
}

// ---------------------------------------------------------------------------
// Launcher
// ---------------------------------------------------------------------------
extern "C" void kernel_launch(void* const* d_in, const int* in_sizes, int n_in,
                              void* d_out, int out_size, void* d_ws, size_t ws_size,
                              hipStream_t stream) {
  const float* x_in       = (const float*)d_in[0];
  const long long* eidx   = (const long long*)d_in[1];
  const float* edge_attr  = (const float*)d_in[2];
  const float* Wq = (const float*)d_in[3];   const float* bq = (const float*)d_in[4];
  const float* Wk = (const float*)d_in[5];   const float* bk = (const float*)d_in[6];
  const float* Wv = (const float*)d_in[7];   const float* bv = (const float*)d_in[8];
  const float* We = (const float*)d_in[9];
  const float* Wsk = (const float*)d_in[10]; const float* bsk = (const float*)d_in[11];
  const float* ee_W1 = (const float*)d_in[12]; const float* ee_b1 = (const float*)d_in[13];
  const float* ee_W2 = (const float*)d_in[14]; const float* ee_b2 = (const float*)d_in[15];
  const float* bn_g = (const float*)d_in[16]; const float* bn_b = (const float*)d_in[17];
  const float* ln_g = (const float*)d_in[18]; const float* ln_b = (const float*)d_in[19];

  const long N = NN, E = EEDG;
  const long long* rowI = eidx;
  const long long* colI = eidx + E;

  float* out_x = (float*)d_out;             // [N,64]
  float* out_e = out_x + N * 64;            // [E,64]

  // workspace layout
  float* w = (float*)d_ws;
  float* ee    = w; w += E * 64;
  float* qb    = w; w += N * 64;
  float* kb2   = w; w += N * 64;
  float* vb    = w; w += N * 64;
  float* xA    = w; w += N * 64;
  float* xB    = w; w += N * 64;
  float* hb    = w; w += N * 64;
  float* alpha = w; w += E * 4;
  float* expa  = w; w += E * 4;
  float* mx    = w; w += N * 4;
  float* ssum  = w; w += N * 4;
  float* rp    = w; w += E;
  float* colsum = w; w += 64;
  float* colsq  = w; w += 64;
  float* rstats = w; w += 2;

  const int TB = 256;
  const int gemmBlocksE = (int)((E + 63) / 64);   // 128 thr = 4 waves * 16 rows
  const int gemmBlocksN = (int)((N + 63) / 64);
  const int blkE   = (int)((E + TB - 1) / TB);
  const int blkE4  = (int)((E * 4 + TB - 1) / TB);
  const int blkE64 = (int)((E * 64 + TB - 1) / TB);
  const int blkN4  = (int)((N * 4 + TB - 1) / TB);

  // 1) edge encoder MLP -> out_e  (fused 2x WMMA GEMM through LDS)
  edge_mlp_kernel<<<gemmBlocksE, 128, 0, stream>>>(edge_attr, ee_W1, ee_b1, ee_W2, ee_b2,
                                                   out_e, E);

  // 2) relative positional scalar, standardized, broadcast-added to e
  hipMemsetAsync(rstats, 0, 2 * sizeof(float), stream);
  rp_kernel<<<blkE, TB, 0, stream>>>(x_in, rowI, colI, rp, rstats, E);
  rp_apply_kernel<<<blkE64, TB, 0, stream>>>(out_e, rp, rstats, E);

  // 3) three TransformerConv layers
  const float* xcur = x_in;
  for (int layer = 0; layer < 3; ++layer) {
    const float* Wq_i = Wq + (long)layer * 64 * 64;  const float* bq_i = bq + (long)layer * 64;
    const float* Wk_i = Wk + (long)layer * 64 * 64;  const float* bk_i = bk + (long)layer * 64;
    const float* Wv_i = Wv + (long)layer * 64 * 64;  const float* bv_i = bv + (long)layer * 64;
    const float* We_i = We + (long)layer * 64 * 64;
    const float* Ws_i = Wsk + (long)layer * 64 * 64; const float* bs_i = bsk + (long)layer * 64;

    float* houtacc = (layer < 2) ? hb : out_x;  // skip-initialized accumulator

    // q/k/v + skip (skip written directly into accumulator)
    qkvs_kernel<<<gemmBlocksN, 128, 0, stream>>>(xcur, Wq_i, bq_i, Wk_i, bk_i, Wv_i, bv_i,
                                                 Ws_i, bs_i, qb, kb2, vb, houtacc, N);
    // ee = e @ We (no bias)
    gemm64_kernel<<<gemmBlocksE, 128, 0, stream>>>(out_e, We_i, (const float*)nullptr, ee, E, 0);

    // segment softmax over destination nodes
    init_softmax_kernel<<<blkN4, TB, 0, stream>>>(mx, ssum, N * 4);
    alpha_kernel<<<blkE4, TB, 0, stream>>>(qb, kb2, ee, rowI, colI, alpha, mx, E);
    expsum_kernel<<<blkE4, TB, 0, stream>>>(alpha, colI, mx, expa, ssum, E);
    scatter_kernel<<<blkE64, TB, 0, stream>>>(vb, ee, expa, ssum, rowI, colI, houtacc, E);

    if (layer < 2) {
      float* xnext = (layer == 0) ? xA : xB;
      hipMemsetAsync(colsum, 0, 128 * sizeof(float), stream);  // colsum+colsq contiguous
      bnstats_kernel<<<512, TB, 0, stream>>>(hb, colsum, colsq, N * 64);
      bnln_kernel<<<(int)N, 64, 0, stream>>>(hb, xcur, colsum, colsq,
                                             bn_g + (long)layer * 64, bn_b + (long)layer * 64,
                                             ln_g + (long)layer * 64, ln_b + (long)layer * 64,
                                             xnext, N);
      xcur = xnext;
    }
  }
  (void)in_sizes; (void)n_in; (void)out_size; (void)ws_size;
}